// InterpLnr_50474455662810
// MI455X (gfx1250) — compile-verified
//
#include <hip/hip_runtime.h>
#include <hip/hip_bf16.h>
#include <stdint.h>

// ---- problem constants (match reference hparams) ----
#define B_          32
#define D_          80
#define TX_         2560        // MAX_LEN_PAD
#define MAXSEG_     108         // MAX_NUM_SEG
#define S_          64
#define N_          (B_ * MAXSEG_)      // 3456
#define NS_         (N_ * S_)           // 221184
#define MINSEG_     19
#define NB_         (NS_ / 256)         // 864 blocks of 256
#define ROWS_TOT_   (B_ * TX_)          // 81920 output rows

// ---------------------------------------------------------------------------
// Shared element math — mirrors the reference's f32 ops exactly:
//   fs = s / (scales_u + 0.5);  fl = floor(fs);  lam = fs - fl
//   mask = (fl < len_seg - 1) && (fl + offset < len_seq[b] - 1)
// ---------------------------------------------------------------------------
__device__ __forceinline__ void elem_compute(int j,
                                             const float* __restrict__ scales_u,
                                             const int*   __restrict__ len_seg_raw,
                                             const int*   __restrict__ len_seq,
                                             const int*   __restrict__ seg_off,
                                             bool& m, float& lam, int& ifl, int& bsrc) {
    int s = j & (S_ - 1);
    int n = j >> 6;             // j / S_
    int b = n / MAXSEG_;
    float scale = scales_u[n] + 0.5f;
    float fs = (float)s / scale;
    float fl = floorf(fs);
    lam = fs - fl;
    int lseg = len_seg_raw[n] + MINSEG_;
    int off  = seg_off[n];
    float idx_org = fl + (float)off;
    bool c1 = fl < ((float)lseg - 1.0f);
    bool c2 = idx_org < (float)(len_seq[b] - 1);
    m = c1 && c2;
    int i = (int)idx_org;                      // idx_org >= 0 always
    i = i > (TX_ - 2) ? (TX_ - 2) : (i < 0 ? 0 : i);
    ifl = i;
    bsrc = b;
}

// ---------------------------------------------------------------------------
// K0: per-batch exclusive cumsum of segment lengths (108 segments per row)
// ---------------------------------------------------------------------------
__global__ void k0_seg_offsets(const int* __restrict__ len_seg_raw,
                               int* __restrict__ seg_off) {
    int b = threadIdx.x;
    if (b >= B_) return;
    int run = 0;
    for (int s = 0; s < MAXSEG_; ++s) {
        seg_off[b * MAXSEG_ + s] = run;
        run += len_seg_raw[b * MAXSEG_ + s] + MINSEG_;
    }
}

// ---------------------------------------------------------------------------
// K1: per-block mask population count (wave32 ballot reduce)
// ---------------------------------------------------------------------------
__global__ void k1_mask_sums(const float* __restrict__ scales_u,
                             const int*   __restrict__ len_seg_raw,
                             const int*   __restrict__ len_seq,
                             const int*   __restrict__ seg_off,
                             int* __restrict__ block_sums) {
    __shared__ int wsum[8];
    int j = blockIdx.x * 256 + threadIdx.x;
    bool m; float lam; int ifl, bsrc;
    elem_compute(j, scales_u, len_seg_raw, len_seq, seg_off, m, lam, ifl, bsrc);
    int lane = threadIdx.x & 31;
    int wid  = threadIdx.x >> 5;
    unsigned bal = (unsigned)__ballot(m);      // wave32: low 32 bits
    if (lane == 0) wsum[wid] = __popc(bal);
    __syncthreads();
    if (threadIdx.x == 0) {
        int s = 0;
        #pragma unroll
        for (int i = 0; i < 8; ++i) s += wsum[i];
        block_sums[blockIdx.x] = s;
    }
}

// ---------------------------------------------------------------------------
// K2: single-block scan of NB_ block sums -> exclusive offsets, total, rows
// ---------------------------------------------------------------------------
__global__ void k2_scan(const int* __restrict__ block_sums,
                        int* __restrict__ block_offs,
                        int* __restrict__ total_p,
                        int* __restrict__ rows_p) {
    __shared__ int sh[1024];
    int tid = threadIdx.x;
    int v = (tid < NB_) ? block_sums[tid] : 0;
    sh[tid] = v;
    __syncthreads();
    #pragma unroll
    for (int off = 1; off < 1024; off <<= 1) {
        int t = (tid >= off) ? sh[tid - off] : 0;
        __syncthreads();
        sh[tid] += t;
        __syncthreads();
    }
    if (tid < NB_) block_offs[tid] = sh[tid] - v;   // exclusive prefix
    if (tid == NB_ - 1) {
        int total = sh[tid];
        *total_p = total;
        *rows_p  = total / B_;
    }
}

// ---------------------------------------------------------------------------
// K3: recompute mask, intra-block exclusive scan, scatter compact records
//     record: src_pack[p] = (batch << 16) | i_fl ;  lam_c[p] = lam
// ---------------------------------------------------------------------------
__global__ void k3_compact(const float* __restrict__ scales_u,
                           const int*   __restrict__ len_seg_raw,
                           const int*   __restrict__ len_seq,
                           const int*   __restrict__ seg_off,
                           const int*   __restrict__ block_offs,
                           int*  __restrict__ src_pack,
                           float* __restrict__ lam_c) {
    __shared__ int wsum[8];
    int j = blockIdx.x * 256 + threadIdx.x;
    bool m; float lam; int ifl, bsrc;
    elem_compute(j, scales_u, len_seg_raw, len_seq, seg_off, m, lam, ifl, bsrc);
    int lane = threadIdx.x & 31;
    int wid  = threadIdx.x >> 5;
    unsigned bal = (unsigned)__ballot(m);
    int lexcl = __popc(bal & ((1u << lane) - 1u));
    if (lane == 0) wsum[wid] = __popc(bal);
    __syncthreads();
    int woff = 0;
    for (int i = 0; i < wid; ++i) woff += wsum[i];
    if (m) {
        int p = block_offs[blockIdx.x] + woff + lexcl;
        src_pack[p] = (bsrc << 16) | ifl;
        lam_c[p]    = lam;
    }
}

// ---------------------------------------------------------------------------
// Issue the 640B contiguous window x[b, i_fl : i_fl+2, :] (40 x 16B chunks)
// into an LDS slot with exactly TWO global_load_async_to_lds_b128 instructions
// (per-lane 16B transfers, tracked by ASYNCcnt).
// ---------------------------------------------------------------------------
__device__ __forceinline__ void issue_row_async(unsigned lds_base,
                                                const float* gsrc, int lane) {
    {
        unsigned la = lds_base + (unsigned)(lane * 16);
        const float* ga = gsrc + lane * 4;
        asm volatile("global_load_async_to_lds_b128 %0, %1, off"
                     :: "v"(la), "v"(ga) : "memory");
    }
    if (lane < 8) {
        unsigned la = lds_base + (unsigned)((32 + lane) * 16);
        const float* ga = gsrc + (32 + lane) * 4;
        asm volatile("global_load_async_to_lds_b128 %0, %1, off"
                     :: "v"(la), "v"(ga) : "memory");
    }
}

// ---------------------------------------------------------------------------
// K4: one wave32 per output row, two-deep pipeline over double-buffered LDS.
//   iter i: issue row r+nwaves into buf^1, s_wait_asynccnt {2|0}, lerp+store
//   buf. Async loads complete in-order, so asynccnt<=2 ==> current row done.
// ---------------------------------------------------------------------------
__global__ void k4_gather_out(const float* __restrict__ x,
                              const int*   __restrict__ src_pack,
                              const float* __restrict__ lam_c,
                              const int*   __restrict__ rows_p,
                              float* __restrict__ out) {
    __shared__ __align__(16) float stage[8][2][160];   // 8 waves x 2 x 640B
    const int lane = threadIdx.x & 31;
    const int wid  = threadIdx.x >> 5;
    const int waves_per_blk = blockDim.x >> 5;
    const int nwaves = gridDim.x * waves_per_blk;
    const int wgid = blockIdx.x * waves_per_blk + wid;
    const int rows = *rows_p;                          // uniform
    unsigned lds_base[2];
    lds_base[0] = (unsigned)(uintptr_t)&stage[wid][0][0];
    lds_base[1] = (unsigned)(uintptr_t)&stage[wid][1][0];

    int r = wgid;
    bool cur_valid = (r < ROWS_TOT_);
    bool cur_data = false;
    float cur_lam = 0.0f;
    float* cur_orow = nullptr;
    int buf = 0;

    // prologue: stage row r into buf 0
    if (cur_valid) {
        int b = r / TX_;
        int t = r - b * TX_;
        cur_orow = out + (size_t)r * D_;
        if (t < rows) {
            int p = b * rows + t;
            int pack = src_pack[p];
            cur_lam = lam_c[p];
            const float* gsrc = x + ((size_t)(pack >> 16) * TX_ + (pack & 0xFFFF)) * D_;
            issue_row_async(lds_base[0], gsrc, lane);
            cur_data = true;
        }
    }

    while (cur_valid) {
        // stage next row into the other buffer before waiting on current
        int rn = r + nwaves;
        bool nxt_valid = (rn < ROWS_TOT_);
        bool nxt_data = false;
        float nxt_lam = 0.0f;
        float* nxt_orow = nullptr;
        if (nxt_valid) {
            int b = rn / TX_;
            int t = rn - b * TX_;
            nxt_orow = out + (size_t)rn * D_;
            if (t < rows) {
                int p = b * rows + t;
                int pack = src_pack[p];
                nxt_lam = lam_c[p];
                const float* gsrc = x + ((size_t)(pack >> 16) * TX_ + (pack & 0xFFFF)) * D_;
                issue_row_async(lds_base[buf ^ 1], gsrc, lane);
                nxt_data = true;
            }
        }

        if (cur_data) {
            if (nxt_data) { asm volatile("s_wait_asynccnt 0x2" ::: "memory"); }
            else          { asm volatile("s_wait_asynccnt 0x0" ::: "memory"); }
            if (lane < 20) {
                const float4 vf = *(const float4*)&stage[wid][buf][lane * 4];
                const float4 vc = *(const float4*)&stage[wid][buf][D_ + lane * 4];
                float4 o;
                o.x = (1.0f - cur_lam) * vf.x + cur_lam * vc.x;
                o.y = (1.0f - cur_lam) * vf.y + cur_lam * vc.y;
                o.z = (1.0f - cur_lam) * vf.z + cur_lam * vc.z;
                o.w = (1.0f - cur_lam) * vf.w + cur_lam * vc.w;
                *(float4*)&cur_orow[lane * 4] = o;
            }
        } else {
            if (lane < 20) {
                float4 z; z.x = 0.0f; z.y = 0.0f; z.z = 0.0f; z.w = 0.0f;
                *(float4*)&cur_orow[lane * 4] = z;
            }
        }

        r = rn;
        cur_valid = nxt_valid;
        cur_data = nxt_data;
        cur_lam = nxt_lam;
        cur_orow = nxt_orow;
        buf ^= 1;
    }
}

// ---------------------------------------------------------------------------
extern "C" void kernel_launch(void* const* d_in, const int* in_sizes, int n_in,
                              void* d_out, int out_size, void* d_ws, size_t ws_size,
                              hipStream_t stream) {
    const float* x           = (const float*)d_in[0];   // (32, 2560, 80)
    const int*   len_seq     = (const int*)  d_in[1];   // (32,)
    const float* scales_u    = (const float*)d_in[2];   // (3456,)
    const int*   len_seg_raw = (const int*)  d_in[3];   // (3456, 1)
    float* out = (float*)d_out;                         // (32, 2560, 80)

    // workspace carve-out (~1.8 MB total, 4B aligned)
    char* w = (char*)d_ws;
    int*   seg_off    = (int*)w;  w += (size_t)N_  * sizeof(int);
    int*   block_sums = (int*)w;  w += (size_t)NB_ * sizeof(int);
    int*   block_offs = (int*)w;  w += (size_t)NB_ * sizeof(int);
    int*   total_p    = (int*)w;  w += sizeof(int);
    int*   rows_p     = (int*)w;  w += sizeof(int);
    int*   src_pack   = (int*)w;  w += (size_t)NS_ * sizeof(int);
    float* lam_c      = (float*)w;

    k0_seg_offsets<<<1, 32, 0, stream>>>(len_seg_raw, seg_off);
    k1_mask_sums  <<<NB_, 256, 0, stream>>>(scales_u, len_seg_raw, len_seq, seg_off, block_sums);
    k2_scan       <<<1, 1024, 0, stream>>>(block_sums, block_offs, total_p, rows_p);
    k3_compact    <<<NB_, 256, 0, stream>>>(scales_u, len_seg_raw, len_seq, seg_off,
                                            block_offs, src_pack, lam_c);
    k4_gather_out <<<512, 256, 0, stream>>>(x, src_pack, lam_c, rows_p, out);
}